// CUDAVoxelizer_49194555408638
// MI455X (gfx1250) — compile-verified
//
#include <hip/hip_runtime.h>
#include <hip/hip_bf16.h>
#include <math.h>

#define NGAUSS 16384
#define FEAT   17
#define GX     200
#define GY     200
#define GZ     16
#define NVOX   (GX * GY * GZ)
#define VOXELF 0.4f
#define SIGMAF 3.0f
#define EPSF   1e-6f
#define VMINX  -40.0f
#define VMINY  -40.0f
#define VMINZ  -1.0f
#define NOFF   343   /* 7*7*7 */

typedef __attribute__((ext_vector_type(2))) float v2f;
typedef __attribute__((ext_vector_type(8))) float v8f;

// One wave (32 lanes) per gaussian. 343 offsets processed in 11 chunks of 32.
// Feature scatter uses V_WMMA_F32_16X16X4_F32 as a 16x16 outer product:
//   A[16x4]: column K=0 = contrib for 16 cells (lane m, VGPR0)
//   B[4x16]: row    K=0 = features[0..15]      (lanes 0-15, VGPR0)
//   D[16x16] = contrib_M * feat_N, scattered via atomics using the C/D layout.
__global__ void __launch_bounds__(256)
splat_kernel(const float* __restrict__ means,
             const float* __restrict__ opac,
             const float* __restrict__ scales,
             const float* __restrict__ rots,
             const float* __restrict__ feats,
             float* __restrict__ out)
{
    const int lane = threadIdx.x & 31;
    const int g    = (blockIdx.x * blockDim.x + threadIdx.x) >> 5;  // wave id == gaussian id

    // ---- per-gaussian (wave-uniform) setup -------------------------------
    const float mx = means[3 * g + 0];
    const float my = means[3 * g + 1];
    const float mz = means[3 * g + 2];
    const float op = opac[g];
    const float sx = scales[3 * g + 0];
    const float sy = scales[3 * g + 1];
    const float sz = scales[3 * g + 2];
    const float qr0 = rots[4 * g + 0];
    const float qx0 = rots[4 * g + 1];
    const float qy0 = rots[4 * g + 2];
    const float qz0 = rots[4 * g + 3];

    const float qinv = 1.0f / sqrtf(qr0 * qr0 + qx0 * qx0 + qy0 * qy0 + qz0 * qz0 + 1e-8f);
    const float r = qr0 * qinv, x = qx0 * qinv, y = qy0 * qinv, z = qz0 * qinv;

    // Rotation matrix rows R[k][i]
    const float R00 = 1.0f - 2.0f * (y * y + z * z);
    const float R01 = 2.0f * (x * y - r * z);
    const float R02 = 2.0f * (x * z + r * y);
    const float R10 = 2.0f * (x * y + r * z);
    const float R11 = 1.0f - 2.0f * (x * x + z * z);
    const float R12 = 2.0f * (y * z - r * x);
    const float R20 = 2.0f * (x * z - r * y);
    const float R21 = 2.0f * (y * z + r * x);
    const float R22 = 1.0f - 2.0f * (x * x + y * y);

    // CovInv = R^T diag(1/s^2) R   (closed form; Cov = R^T diag(s^2) R)
    const float i0 = 1.0f / (sx * sx);
    const float i1 = 1.0f / (sy * sy);
    const float i2 = 1.0f / (sz * sz);
    const float cxx = R00 * R00 * i0 + R10 * R10 * i1 + R20 * R20 * i2;
    const float cyy = R01 * R01 * i0 + R11 * R11 * i1 + R21 * R21 * i2;
    const float czz = R02 * R02 * i0 + R12 * R12 * i1 + R22 * R22 * i2;
    const float cxy = R00 * R01 * i0 + R10 * R11 * i1 + R20 * R21 * i2;
    const float cyz = R01 * R02 * i0 + R11 * R12 * i1 + R21 * R22 * i2;
    const float cxz = R00 * R02 * i0 + R10 * R12 * i1 + R20 * R22 * i2;

    float smax = sx > sy ? sx : sy; smax = smax > sz ? smax : sz;
    int radii = (int)ceilf(smax * SIGMAF / VOXELF);
    if (radii < 1) radii = 1;

    const int ix = (int)floorf((mx - VMINX) / VOXELF);
    const int iy = (int)floorf((my - VMINY) / VOXELF);
    const int iz = (int)floorf((mz - VMINZ) / VOXELF);

    // features: lane n (0..15) holds feature n for the WMMA B operand
    const float flane = (lane < FEAT) ? feats[g * FEAT + lane] : 0.0f;
    const float f16v  = feats[g * FEAT + 16];
    v2f bmat;
    bmat[0] = (lane < 16) ? flane : 0.0f;
    bmat[1] = 0.0f;

    const int col   = lane & 15;
    const int rbase = (lane >= 16) ? 8 : 0;

    // ---- 11 uniform chunks of 32 offsets (343 total) ---------------------
    #pragma unroll 1
    for (int c = 0; c < 11; ++c) {
        const int o  = c * 32 + lane;
        const int oo = (o < NOFF) ? o : 0;
        const int oi = oo / 49 - 3;
        const int oj = (oo / 7) % 7 - 3;
        const int ok = oo % 7 - 3;

        const int vx = ix + oi, vy = iy + oj, vz = iz + ok;
        const float dxx = mx - ((float)vx * VOXELF + VMINX + 0.5f * VOXELF);
        const float dyy = my - ((float)vy * VOXELF + VMINY + 0.5f * VOXELF);
        const float dzz = mz - ((float)vz * VOXELF + VMINZ + 0.5f * VOXELF);

        const float power = -0.5f * (cxx * dxx * dxx + cyy * dyy * dyy + czz * dzz * dzz)
                            - (cxy * dxx * dyy + cyz * dyy * dzz + cxz * dxx * dzz);
        float contrib = op * expf(power);

        const int aoi = oi < 0 ? -oi : oi;
        const int aoj = oj < 0 ? -oj : oj;
        const int aok = ok < 0 ? -ok : ok;
        const bool in_r = (aoi <= radii) && (aoj <= radii) && (aok <= radii);
        const bool in_g = (vx >= 0) && (vx < GX) && (vy >= 0) && (vy < GY) && (vz >= 0) && (vz < GZ);
        const bool valid = (o < NOFF) && in_r && in_g;
        contrib = valid ? contrib : 0.0f;

        int cxi = vx < 0 ? 0 : (vx >= GX ? GX - 1 : vx);
        int cyi = vy < 0 ? 0 : (vy >= GY ? GY - 1 : vy);
        int czi = vz < 0 ? 0 : (vz >= GZ ? GZ - 1 : vz);
        const int flat = cxi * (GY * GZ) + cyi * GZ + czi;

        // density + 17th feature channel (scalar path)
        if (contrib != 0.0f) {
            atomicAdd(&out[flat], contrib);
            atomicAdd(&out[NVOX + flat * FEAT + 16], contrib * f16v);
        }

        // ---- WMMA outer products: contrib (16 rows) x features (16 cols) --
        const float cHiSrc = __shfl(contrib, (lane & 15) + 16, 32);
        v2f a0, a1;
        a0[0] = (lane < 16) ? contrib : 0.0f;  a0[1] = 0.0f;
        a1[0] = (lane < 16) ? cHiSrc  : 0.0f;  a1[1] = 0.0f;
        v8f d0 = {};
        v8f d1 = {};
        d0 = __builtin_amdgcn_wmma_f32_16x16x4_f32(false, a0, false, bmat,
                                                   (short)0, d0, false, false);
        d1 = __builtin_amdgcn_wmma_f32_16x16x4_f32(false, a1, false, bmat,
                                                   (short)0, d1, false, false);

        // Scatter: lane holds column N = lane&15; VGPR r = row M (rbase+r).
        #pragma unroll
        for (int rr = 0; rr < 8; ++rr) {
            const int row  = rbase + rr;
            const int f0   = __shfl(flat, row, 32);
            const float v0 = d0[rr];
            if (v0 != 0.0f) atomicAdd(&out[NVOX + f0 * FEAT + col], v0);
            const int f1   = __shfl(flat, row + 16, 32);
            const float v1 = d1[rr];
            if (v1 != 0.0f) atomicAdd(&out[NVOX + f1 * FEAT + col], v1);
        }
    }
}

__global__ void normalize_kernel(float* __restrict__ out)
{
    const int v = blockIdx.x * blockDim.x + threadIdx.x;
    if (v < NVOX) {
        const float den = fmaxf(out[v], EPSF);
        float* f = &out[NVOX + v * FEAT];
        #pragma unroll
        for (int j = 0; j < FEAT; ++j) f[j] = f[j] / den;
    }
}

extern "C" void kernel_launch(void* const* d_in, const int* in_sizes, int n_in,
                              void* d_out, int out_size, void* d_ws, size_t ws_size,
                              hipStream_t stream)
{
    const float* means  = (const float*)d_in[0];
    const float* opac   = (const float*)d_in[1];
    const float* scales = (const float*)d_in[2];
    const float* rots   = (const float*)d_in[3];
    const float* feats  = (const float*)d_in[4];
    float* out = (float*)d_out;

    // zero accumulators (density + features) every call: graph-replay safe
    hipMemsetAsync(out, 0, (size_t)NVOX * (1 + FEAT) * sizeof(float), stream);

    // 16384 gaussians, one wave32 each, 8 waves per block
    splat_kernel<<<dim3(NGAUSS / 8), dim3(256), 0, stream>>>(means, opac, scales,
                                                             rots, feats, out);

    normalize_kernel<<<dim3((NVOX + 255) / 256), dim3(256), 0, stream>>>(out);
}